// SGCLayer_30683246363240
// MI455X (gfx1250) — compile-verified
//
#include <hip/hip_runtime.h>
#include <hip/hip_bf16.h>

typedef __attribute__((ext_vector_type(2))) float v2f;
typedef __attribute__((ext_vector_type(8))) float v8f;

#define TPB 256

struct SelState {
    unsigned prefix;   // accumulated high bits of threshold key
    unsigned k;        // remaining rank within candidates
    unsigned thresh;   // final threshold key (key of K-th smallest)
    unsigned ktie;     // # ties (== thresh) to include, in index order
};

// ---- monotonic uint encoding of float (ascending float -> ascending uint) ----
__device__ __forceinline__ unsigned f2ord(float f) {
    unsigned u = __float_as_uint(f);
    return (u & 0x80000000u) ? ~u : (u | 0x80000000u);
}

// ---------------------------------------------------------------- utility ----
__global__ void zero_u32_kernel(unsigned* p, long n) {
    long i = (long)blockIdx.x * blockDim.x + threadIdx.x;
    long stride = (long)gridDim.x * blockDim.x;
    for (; i < n; i += stride) p[i] = 0u;
}

__global__ void degree_kernel(const int* __restrict__ dst, unsigned* __restrict__ deg, int E) {
    int e = blockIdx.x * blockDim.x + threadIdx.x;
    if (e < E) atomicAdd(&deg[dst[e]], 1u);
}

__global__ void norm_kernel(const unsigned* __restrict__ deg, float* __restrict__ norm, int N) {
    int i = blockIdx.x * blockDim.x + threadIdx.x;
    if (i < N) {
        float d = (float)(deg[i] < 1u ? 1u : deg[i]);
        norm[i] = rsqrtf(d);
    }
}

// one thread per node: 1 / max(||h_row||, 1e-12)
__global__ void rownorm_kernel(const float* __restrict__ h, float* __restrict__ rnorm, int N) {
    int i = blockIdx.x * blockDim.x + threadIdx.x;
    if (i >= N) return;
    const float4* row = (const float4*)(h + (size_t)i * 32);
    float s = 0.f;
#pragma unroll
    for (int j = 0; j < 8; ++j) {
        float4 v = row[j];
        s += v.x * v.x + v.y * v.y + v.z * v.z + v.w * v.w;
    }
    float nrm = sqrtf(s);
    rnorm[i] = 1.0f / fmaxf(nrm, 1e-12f);
}

// per-edge cosine similarity -> ordered uint key (L2-resident gathers)
__global__ void coskey_kernel(const float* __restrict__ h, const float* __restrict__ rnorm,
                              const int* __restrict__ src, const int* __restrict__ dst,
                              unsigned* __restrict__ keys, int E) {
    int e = blockIdx.x * blockDim.x + threadIdx.x;
    if (e >= E) return;
    int s = src[e], d = dst[e];
    const float4* ps = (const float4*)(h + (size_t)s * 32);
    const float4* pd = (const float4*)(h + (size_t)d * 32);
    float acc = 0.f;
#pragma unroll
    for (int j = 0; j < 8; ++j) {
        float4 a = ps[j], b = pd[j];
        acc += a.x * b.x + a.y * b.y + a.z * b.z + a.w * b.w;
    }
    float c = acc * rnorm[s] * rnorm[d];
    keys[e] = f2ord(c);
}

// ---------------------------------------------------------- radix select ----
__global__ void selinit_kernel(SelState* st, unsigned* hist, unsigned K) {
    if (threadIdx.x == 0) { st->prefix = 0u; st->k = K; st->thresh = 0u; st->ktie = 0u; }
    hist[threadIdx.x] = 0u;   // blockDim == 256
}

__global__ void hist_kernel(const unsigned* __restrict__ keys, const SelState* __restrict__ st,
                            unsigned* __restrict__ hist, int E, int shift, unsigned maskHigh) {
    __shared__ unsigned lh[256];
    lh[threadIdx.x] = 0u;
    __syncthreads();
    unsigned prefix = st->prefix;
    long i = (long)blockIdx.x * blockDim.x + threadIdx.x;
    long stride = (long)gridDim.x * blockDim.x;
    for (; i < E; i += stride) {
        unsigned k = keys[i];
        if (((k ^ prefix) & maskHigh) == 0u)
            atomicAdd(&lh[(k >> shift) & 255u], 1u);
    }
    __syncthreads();
    unsigned v = lh[threadIdx.x];
    if (v) atomicAdd(&hist[threadIdx.x], v);
}

// single-thread digit pick; also re-zeroes hist for next pass
__global__ void select_kernel(unsigned* hist, SelState* st, int shift, int lastPass) {
    unsigned k = st->k;
    unsigned cum = 0u, digit = 255u;
    for (unsigned d = 0; d < 256u; ++d) {
        unsigned c = hist[d];
        if (cum + c >= k) { digit = d; break; }
        cum += c;
    }
    st->prefix |= (digit << shift);
    st->k = k - cum;
    if (lastPass) { st->thresh = st->prefix; st->ktie = st->k; }
    for (unsigned d = 0; d < 256u; ++d) hist[d] = 0u;
}

// ---------------------------------------------------- edge mask w/ ties ----
// pass1: w = 0 (key<T), 1 (key>T), 2 (tie, resolve later); per-block tie counts
__global__ void wpass1_kernel(const unsigned* __restrict__ keys, const SelState* __restrict__ st,
                              unsigned char* __restrict__ w, unsigned* __restrict__ blockTie, int E) {
    __shared__ unsigned cnt;
    if (threadIdx.x == 0) cnt = 0u;
    __syncthreads();
    int e = blockIdx.x * blockDim.x + threadIdx.x;
    if (e < E) {
        unsigned k = keys[e], T = st->thresh;
        unsigned char ww;
        if (k < T) ww = 0;
        else if (k > T) ww = 1;
        else { ww = 2; atomicAdd(&cnt, 1u); }
        w[e] = ww;
    }
    __syncthreads();
    if (threadIdx.x == 0) blockTie[blockIdx.x] = cnt;
}

__global__ void tiescan_kernel(const unsigned* __restrict__ blockTie, unsigned* __restrict__ blockOff, int nb) {
    unsigned run = 0u;
    for (int b = 0; b < nb; ++b) { blockOff[b] = run; run += blockTie[b]; }
}

// one thread per 256-edge block: resolve ties in global index order
__global__ void wpass2_kernel(unsigned char* __restrict__ w, const unsigned* __restrict__ blockOff,
                              const SelState* __restrict__ st, int E) {
    int b = blockIdx.x;
    unsigned rank = blockOff[b];
    unsigned ktie = st->ktie;
    int base = b * 256;
    for (int j = 0; j < 256; ++j) {
        int e = base + j;
        if (e >= E) break;
        if (w[e] == 2) { w[e] = (rank < ktie) ? 0 : 1; rank++; }
    }
}

// ---------------------------------------------------------- propagation ----
// 8 threads per edge, 4 floats each; f32 L2 atomics
__global__ void scatter_kernel(const float* __restrict__ h, const float* __restrict__ norm,
                               const int* __restrict__ src, const int* __restrict__ dst,
                               const unsigned char* __restrict__ w, float* __restrict__ hn, int E) {
    long t = (long)blockIdx.x * blockDim.x + threadIdx.x;
    int e = (int)(t >> 3);
    int c = (int)(t & 7);
    if (e >= E) return;
    if (!w[e]) return;
    int s = src[e], d = dst[e];
    float ns = norm[s];
    float4 v = ((const float4*)(h + (size_t)s * 32))[c];
    float* o = hn + (size_t)d * 32 + c * 4;
    atomicAdd(o + 0, v.x * ns);
    atomicAdd(o + 1, v.y * ns);
    atomicAdd(o + 2, v.z * ns);
    atomicAdd(o + 3, v.w * ns);
}

__global__ void scale_kernel(float* __restrict__ h, const float* __restrict__ norm, int N) {
    long i = (long)blockIdx.x * blockDim.x + threadIdx.x;
    if (i >= (long)N * 32) return;
    h[i] *= norm[i >> 5];
}

// ------------------------------------------------------------ final GEMM ----
// out[N,64] = h[N,32] @ W[64,32]^T via chained V_WMMA_F32_16X16X4_F32.
// One wave computes a 16x64 output tile: 8 K-steps x 4 N-tiles = 32 WMMAs.
// A layout (16x4 f32): lanes 0-15 hold K={0,1} in v[0],v[1]; lanes 16-31 K={2,3}.
// B layout (4x16 f32): lane -> column n, lanes 0-15 K={0,1}, lanes 16-31 K={2,3}.
// C/D (16x16 f32): lane 0-15 n=lane / M=r; lanes 16-31 n=lane-16 / M=8+r.
__global__ void gemm_wmma_kernel(const float* __restrict__ h, const float* __restrict__ W,
                                 float* __restrict__ out, int ntiles) {
    int tile = blockIdx.x * (blockDim.x >> 5) + (threadIdx.x >> 5);   // wave-uniform
    if (tile >= ntiles) return;                                       // whole wave exits: EXEC stays all-1s
    int lane = threadIdx.x & 31;
    int half = lane >> 4;     // 0: K pair {0,1}; 1: K pair {2,3}
    int l15 = lane & 15;

    const float* arow = h + (size_t)(tile * 16 + l15) * 32 + half * 2;
    const float* wrow = W + (size_t)l15 * 32 + half * 2;

    v8f acc0 = {}, acc1 = {}, acc2 = {}, acc3 = {};
#pragma unroll
    for (int ks = 0; ks < 8; ++ks) {
        v2f a; a.x = arow[ks * 4 + 0]; a.y = arow[ks * 4 + 1];
        v2f b0, b1, b2, b3;
        const float* wp = wrow + ks * 4;
        b0.x = wp[0];           b0.y = wp[1];
        b1.x = wp[16 * 32];     b1.y = wp[16 * 32 + 1];
        b2.x = wp[32 * 32];     b2.y = wp[32 * 32 + 1];
        b3.x = wp[48 * 32];     b3.y = wp[48 * 32 + 1];
        acc0 = __builtin_amdgcn_wmma_f32_16x16x4_f32(false, a, false, b0, (short)0, acc0, false, false);
        acc1 = __builtin_amdgcn_wmma_f32_16x16x4_f32(false, a, false, b1, (short)0, acc1, false, false);
        acc2 = __builtin_amdgcn_wmma_f32_16x16x4_f32(false, a, false, b2, (short)0, acc2, false, false);
        acc3 = __builtin_amdgcn_wmma_f32_16x16x4_f32(false, a, false, b3, (short)0, acc3, false, false);
    }
#pragma unroll
    for (int r = 0; r < 8; ++r) {
        size_t mm = (size_t)(tile * 16 + r + 8 * half);
        out[mm * 64 +  0 + l15] = acc0[r];
        out[mm * 64 + 16 + l15] = acc1[r];
        out[mm * 64 + 32 + l15] = acc2[r];
        out[mm * 64 + 48 + l15] = acc3[r];
    }
}

// ----------------------------------------------------------------- host ----
extern "C" void kernel_launch(void* const* d_in, const int* in_sizes, int n_in,
                              void* d_out, int out_size, void* d_ws, size_t ws_size,
                              hipStream_t stream) {
    const float* features = (const float*)d_in[0];
    const int*   src      = (const int*)d_in[1];
    const int*   dst      = (const int*)d_in[2];
    const float* W        = (const float*)d_in[3];

    const int DIN  = 32;
    const int N    = in_sizes[0] / DIN;
    const int E    = in_sizes[1];
    const unsigned KCUT = (unsigned)(E / 10);
    const int nb   = (E + TPB - 1) / TPB;   // 256-edge blocks for tie scan

    // carve workspace (256B aligned slabs)
    char* ws = (char*)d_ws;
    auto carve = [&](size_t bytes) -> void* {
        void* p = (void*)ws;
        ws += (bytes + 255) & ~(size_t)255;
        return p;
    };
    float*         norm     = (float*)carve((size_t)N * 4);
    float*         rnorm    = (float*)carve((size_t)N * 4);
    float*         hA       = (float*)carve((size_t)N * DIN * 4);
    float*         hB       = (float*)carve((size_t)N * DIN * 4);
    unsigned*      keys     = (unsigned*)carve((size_t)E * 4);
    unsigned char* wmask    = (unsigned char*)carve((size_t)E);
    unsigned*      deg      = (unsigned*)carve((size_t)N * 4);
    unsigned*      hist     = (unsigned*)carve(256 * 4);
    SelState*      st       = (SelState*)carve(sizeof(SelState));
    unsigned*      blockTie = (unsigned*)carve((size_t)nb * 4);
    unsigned*      blockOff = (unsigned*)carve((size_t)nb * 4);

    const int gN   = (N + TPB - 1) / TPB;
    const int gE   = (E + TPB - 1) / TPB;
    const int gNE  = ((int)((long)N * DIN) + TPB - 1) / TPB;
    const int gSC  = (int)(((long)E * 8 + TPB - 1) / TPB);

    // degree + symmetric norm
    zero_u32_kernel<<<gN, TPB, 0, stream>>>(deg, (long)N);
    degree_kernel<<<gE, TPB, 0, stream>>>(dst, deg, E);
    norm_kernel<<<gN, TPB, 0, stream>>>(deg, norm, N);

    // h <- features
    hipMemcpyAsync(hA, features, (size_t)N * DIN * 4, hipMemcpyDeviceToDevice, stream);

    float* hcur = hA;
    float* hnxt = hB;
    for (int hop = 0; hop < 2; ++hop) {
        rownorm_kernel<<<gN, TPB, 0, stream>>>(hcur, rnorm, N);
        coskey_kernel<<<gE, TPB, 0, stream>>>(hcur, rnorm, src, dst, keys, E);

        // exact bottom-KCUT via 4-pass radix select on ordered keys
        selinit_kernel<<<1, 256, 0, stream>>>(st, hist, KCUT);
        for (int p = 0; p < 4; ++p) {
            int shift = 24 - 8 * p;
            unsigned maskHigh = (p == 0) ? 0u : (0xFFFFFFFFu << (shift + 8));
            hist_kernel<<<1024, 256, 0, stream>>>(keys, st, hist, E, shift, maskHigh);
            select_kernel<<<1, 1, 0, stream>>>(hist, st, shift, (p == 3) ? 1 : 0);
        }
        wpass1_kernel<<<nb, TPB, 0, stream>>>(keys, st, wmask, blockTie, E);
        tiescan_kernel<<<1, 1, 0, stream>>>(blockTie, blockOff, nb);
        wpass2_kernel<<<nb, 1, 0, stream>>>(wmask, blockOff, st, E);

        // D^-1/2 A D^-1/2 with pruned edges
        zero_u32_kernel<<<gNE, TPB, 0, stream>>>((unsigned*)hnxt, (long)N * DIN);
        scatter_kernel<<<gSC, TPB, 0, stream>>>(hcur, norm, src, dst, wmask, hnxt, E);
        scale_kernel<<<gNE, TPB, 0, stream>>>(hnxt, norm, N);

        float* t = hcur; hcur = hnxt; hnxt = t;
    }

    // fc: out = h @ W^T  (N = 6250 * 16 exactly)
    int ntiles = N / 16;                      // 16-row tiles
    int wavesPerBlock = TPB / 32;             // 8
    int gG = (ntiles + wavesPerBlock - 1) / wavesPerBlock;
    gemm_wmma_kernel<<<gG, TPB, 0, stream>>>(hcur, W, (float*)d_out, ntiles);
}